// SSIMLoss_58342835749680
// MI455X (gfx1250) — compile-verified
//
#include <hip/hip_runtime.h>

// ---------------------------------------------------------------------------
// SSIM loss (window=11, sigma=1.5) on [16,3,512,512] fp32 via CDNA5 WMMA.
// Separable Gaussian conv expressed as banded 16x16x4 FP32 WMMA matmuls.
// ---------------------------------------------------------------------------

typedef __attribute__((ext_vector_type(2))) float v2f;
typedef __attribute__((ext_vector_type(8))) float v8f;

#define WMMA_F32X4(A, B, C) \
  __builtin_amdgcn_wmma_f32_16x16x4_f32(false, (A), false, (B), (short)0, (C), false, false)

// Normalized Gaussian, ws=11, sigma=1.5 (symmetric, sums to 1).
__constant__ float GK11[11] = {
    0.00102838f, 0.00759876f, 0.03600077f, 0.10936069f, 0.21300553f,
    0.26601172f,
    0.21300553f, 0.10936069f, 0.03600077f, 0.00759876f, 0.00102838f};

// Per-wave LDS region: patchP[32*28] + patchT[32*28] + H[32*16] = 2304 floats.
#define PATCH_STRIDE 28
#define WAVE_LDS     2304
#define WAVES_PER_BLK 4
#define NTILES       49152   // 48 channel-images * (512/16)^2
#define NBLOCKS      (NTILES / WAVES_PER_BLK)   // 12288

// MODE: 0=p, 1=t, 2=p*p, 3=t*t, 4=p*t
template <int MODE>
__device__ __forceinline__ v8f conv_tile(const float* __restrict__ pP,
                                         const float* __restrict__ pT,
                                         float* __restrict__ Hl,
                                         const v2f* __restrict__ gf,
                                         int j, int kk) {
  // ---- horizontal pass: H = X @ G  (two 16-row groups, K = 28 in 7 chunks)
#pragma unroll
  for (int rg = 0; rg < 2; ++rg) {
    v8f h = {0.f, 0.f, 0.f, 0.f, 0.f, 0.f, 0.f, 0.f};
    const float* rp = pP + (rg * 16 + j) * PATCH_STRIDE;  // A row = lane&15
    const float* rt = pT + (rg * 16 + j) * PATCH_STRIDE;
#pragma unroll
    for (int c = 0; c < 7; ++c) {
      const int col = 4 * c + kk;  // A: K = 4c + (lane>=16 ? 2 : 0) + {0,1}
      v2f a;
      if (MODE == 0) {
        a.x = rp[col]; a.y = rp[col + 1];
      } else if (MODE == 1) {
        a.x = rt[col]; a.y = rt[col + 1];
      } else if (MODE == 2) {
        float x0 = rp[col], x1 = rp[col + 1];
        a.x = x0 * x0; a.y = x1 * x1;
      } else if (MODE == 3) {
        float x0 = rt[col], x1 = rt[col + 1];
        a.x = x0 * x0; a.y = x1 * x1;
      } else {
        a.x = rp[col] * rt[col]; a.y = rp[col + 1] * rt[col + 1];
      }
      h = WMMA_F32X4(a, gf[c], h);
    }
    // C/D layout: vgpr q -> row q (lanes 0-15), row q+8 (lanes 16-31)
    const int rbase = rg * 16 + ((kk >> 1) << 3);
#pragma unroll
    for (int q = 0; q < 8; ++q) Hl[(rbase + q) * 16 + j] = h[q];
  }
  __syncthreads();

  // ---- vertical pass: V = G^T @ H (A fragments of G^T == gf[] by symmetry)
  v8f v = {0.f, 0.f, 0.f, 0.f, 0.f, 0.f, 0.f, 0.f};
#pragma unroll
  for (int c = 0; c < 7; ++c) {
    const int r0 = 4 * c + kk;
    v2f b;
    b.x = Hl[r0 * 16 + j];
    b.y = Hl[(r0 + 1) * 16 + j];
    v = WMMA_F32X4(gf[c], b, v);
  }
  __syncthreads();
  return v;
}

__global__ __launch_bounds__(128) void ssim_stage1(
    const float* __restrict__ pred, const float* __restrict__ target,
    float* __restrict__ partial) {
  __shared__ float smem[WAVES_PER_BLK * WAVE_LDS + 16 + WAVES_PER_BLK];

  const int tid  = threadIdx.x;
  const int lane = tid & 31;
  const int wid  = tid >> 5;
  const int j    = lane & 15;          // N (B/C) or M (A) index
  const int kk   = (lane >> 4) << 1;   // +2 K offset for high half-wave

  float* gpad = smem + WAVES_PER_BLK * WAVE_LDS;  // 16-entry zero-padded kernel
  float* wsum = gpad + 16;
  if (tid < 16) gpad[tid] = (tid < 11) ? GK11[tid] : 0.0f;

  float* pP = smem + wid * WAVE_LDS;   // 32 x 28, zero-padded
  float* pT = pP + 32 * PATCH_STRIDE;  // 32 x 28
  float* Hl = pT + 32 * PATCH_STRIDE;  // 32 x 16 intermediate

  // One 16x16 output tile per wave.
  const int tile = blockIdx.x * WAVES_PER_BLK + wid;
  const int img  = tile >> 10;                 // 0..47
  const int t2   = tile & 1023;
  const int ty   = (t2 >> 5) << 4;             // tile y origin
  const int tx   = (t2 & 31) << 4;             // tile x origin
  const float* P = pred + img * 262144;
  const float* T = target + img * 262144;

  // Fill zero-padded 26x26 halo patch (stored 32x28, rest zeroed).
  for (int idx = lane; idx < 32 * PATCH_STRIDE; idx += 32) {
    const int r = idx / PATCH_STRIDE;
    const int c = idx - r * PATCH_STRIDE;
    const int gy = ty + r - 5;
    const int gx = tx + c - 5;
    float vp = 0.f, vt = 0.f;
    if (r < 26 && c < 26 && (unsigned)gy < 512u && (unsigned)gx < 512u) {
      const int o = (gy << 9) + gx;
      vp = P[o];
      vt = T[o];
    }
    pP[idx] = vp;
    pT[idx] = vt;
  }
  __syncthreads();

  // Banded-Gaussian fragments: gf[c] serves as B (horiz) and A (vert).
  v2f gf[7];
#pragma unroll
  for (int c = 0; c < 7; ++c) {
    const int i0 = 4 * c + kk;
    const int d0 = i0 - j, d1 = i0 + 1 - j;
    gf[c].x = ((unsigned)d0 < 16u) ? gpad[d0] : 0.f;
    gf[c].y = ((unsigned)d1 < 16u) ? gpad[d1] : 0.f;
  }

  const v8f mp  = conv_tile<0>(pP, pT, Hl, gf, j, kk);
  const v8f mt  = conv_tile<1>(pP, pT, Hl, gf, j, kk);
  const v8f cpp = conv_tile<2>(pP, pT, Hl, gf, j, kk);
  const v8f ctt = conv_tile<3>(pP, pT, Hl, gf, j, kk);
  const v8f cpt = conv_tile<4>(pP, pT, Hl, gf, j, kk);

  const float C1 = 1e-4f, C2 = 9e-4f;
  float acc = 0.f;
#pragma unroll
  for (int q = 0; q < 8; ++q) {
    const float a = mp[q], b = mt[q];
    const float spp = cpp[q] - a * a;
    const float stt = ctt[q] - b * b;
    const float spt = cpt[q] - a * b;
    const float num = (2.f * a * b + C1) * (2.f * spt + C2);
    const float den = (a * a + b * b + C1) * (spp + stt + C2);
    acc += num / den;
  }
#pragma unroll
  for (int o = 16; o > 0; o >>= 1) acc += __shfl_xor(acc, o, 32);
  if (lane == 0) wsum[wid] = acc;
  __syncthreads();
  if (tid == 0)
    partial[blockIdx.x] = (wsum[0] + wsum[1]) + (wsum[2] + wsum[3]);
}

__global__ __launch_bounds__(256) void ssim_finalize(
    const float* __restrict__ partial, int n, float* __restrict__ out) {
  __shared__ double sd[256];
  double s = 0.0;
  for (int i = threadIdx.x; i < n; i += 256) s += (double)partial[i];
  sd[threadIdx.x] = s;
  __syncthreads();
  for (int o = 128; o > 0; o >>= 1) {
    if (threadIdx.x < (unsigned)o) sd[threadIdx.x] += sd[threadIdx.x + o];
    __syncthreads();
  }
  if (threadIdx.x == 0)
    out[0] = (float)(1.0 - sd[0] / 12582912.0);  // N = 16*3*512*512
}

extern "C" void kernel_launch(void* const* d_in, const int* in_sizes, int n_in,
                              void* d_out, int out_size, void* d_ws,
                              size_t ws_size, hipStream_t stream) {
  (void)in_sizes; (void)n_in; (void)out_size; (void)ws_size;
  const float* pred   = (const float*)d_in[0];
  const float* target = (const float*)d_in[1];
  float* partial      = (float*)d_ws;  // NBLOCKS floats (48 KB)

  ssim_stage1<<<dim3(NBLOCKS), dim3(128), 0, stream>>>(pred, target, partial);
  ssim_finalize<<<dim3(1), dim3(256), 0, stream>>>(partial, NBLOCKS,
                                                   (float*)d_out);
}